// Blur_Downsample_72112500899882
// MI455X (gfx1250) — compile-verified
//
#include <hip/hip_runtime.h>
#include <hip/hip_bf16.h>

// Blur+downsample == 2x2 weighted average pool, stride 2, fp32 NHWC.
// Memory-bound (AI ~0.2 FLOP/B): use CDNA5 async global->LDS b128 copies
// (ASYNCcnt path) to stage contiguous row quads, then compute from LDS.

typedef int v4i __attribute__((vector_size(16)));
typedef __attribute__((address_space(1))) v4i* gbl_v4i_p;   // global (device) AS
typedef __attribute__((address_space(3))) v4i* lds_v4i_p;   // LDS AS

__device__ __forceinline__ void async_copy_b128(const void* g, void* l) {
#if __has_builtin(__builtin_amdgcn_global_load_async_to_lds_b128)
    void* gnc = const_cast<void*>(g);
    __builtin_amdgcn_global_load_async_to_lds_b128((gbl_v4i_p)gnc, (lds_v4i_p)l,
                                                   /*offset=*/0, /*cpol=*/0);
#else
    asm volatile("global_load_async_to_lds_b128 %0, %1, off"
                 :
                 : "v"((lds_v4i_p)l), "v"(g)
                 : "memory");
#endif
}

__device__ __forceinline__ void wait_async_zero() {
#if __has_builtin(__builtin_amdgcn_s_wait_asynccnt)
    __builtin_amdgcn_s_wait_asynccnt(0);
#else
    asm volatile("s_wait_asynccnt 0" ::: "memory");
#endif
}

// Fixed problem geometry from the reference.
constexpr int W    = 512;         // input width
constexpr int C    = 3;           // channels
constexpr int RF   = W * C;       // 1536 floats per input row
constexpr int ORF  = (W / 2) * C; // 768 floats per output row
constexpr int ROWS = 4;           // input rows staged per block (-> 2 output rows)
constexpr int TILE_F = RF * ROWS; // 6144 floats = 24576 bytes of LDS

__global__ __launch_bounds__(256)
void Blur_Downsample_72112500899882_kernel(const float* __restrict__ x,
                                           const float* __restrict__ k,
                                           float* __restrict__ out) {
    __shared__ float tile[TILE_F];

    const int tid = threadIdx.x;          // 0..255
    const int q   = blockIdx.x;           // row-quad index over all images
    const int b   = q >> 7;               // image  (128 quads per 512-row image)
    const int r   = q & 127;              // quad within image -> input rows 4r..4r+3

    // 4 input rows are contiguous: one 24576-byte linear chunk.
    const float* gsrc = x + ((size_t)b * W + (size_t)ROWS * r) * RF;

    // Stage via 6 async b128 copies: 256 lanes * 16B * 6 = 24576 B.
#pragma unroll
    for (int it = 0; it < (TILE_F / 4) / 256; ++it) {
        const int fo = (it * 256 + tid) * 4;      // float offset, 16B per lane
        async_copy_b128(gsrc + fo, &tile[fo]);
    }

    // 2x2 kernel taps (each 0.25 for this layer, but read the real input).
    const float k00 = k[0], k01 = k[1], k10 = k[2], k11 = k[3];

    wait_async_zero();      // this wave's async transfers landed in LDS
    __syncthreads();        // make all waves' transfers visible

    // Thread t -> output column t of both output rows of this quad.
    // Disjoint LDS reads: 6 consecutive floats per input row per thread.
    const float* r0 = &tile[6 * tid];
    const float* r1 = r0 + RF;
    const float* r2 = r0 + 2 * RF;
    const float* r3 = r0 + 3 * RF;

    const size_t obase = (((size_t)b * 256 + (size_t)2 * r) * 256 + tid) * C;
    float* o0 = out + obase;        // output row 2r
    float* o1 = o0 + ORF;           // output row 2r+1

#pragma unroll
    for (int c = 0; c < C; ++c) {
        o0[c] = k00 * r0[c] + k01 * r0[c + 3] + k10 * r1[c] + k11 * r1[c + 3];
        o1[c] = k00 * r2[c] + k01 * r2[c + 3] + k10 * r3[c] + k11 * r3[c + 3];
    }
}

extern "C" void kernel_launch(void* const* d_in, const int* in_sizes, int n_in,
                              void* d_out, int out_size, void* d_ws, size_t ws_size,
                              hipStream_t stream) {
    const float* x   = (const float*)d_in[0];   // (B,512,512,3) fp32
    const float* k   = (const float*)d_in[1];   // (2,2) fp32
    float*       out = (float*)d_out;           // (B,256,256,3) fp32

    // One block per 4-input-row quad; derive count from input size so any
    // batch size with H=W=512, C=3 works (reference: B=32 -> 4096 blocks).
    const int quads = in_sizes[0] / (ROWS * RF);
    Blur_Downsample_72112500899882_kernel<<<dim3(quads), dim3(256), 0, stream>>>(x, k, out);
}